// SpatialConvLayer_36696200577451
// MI455X (gfx1250) — compile-verified
//
#include <hip/hip_runtime.h>

// Problem constants (fixed by the reference)
#define BDIM    4
#define NNODES  10000
#define CH      32
#define TDIM    8
#define NEDGE   160000
#define BT      32                 // B*T independent slices
#define ROWS    (BT * NNODES)      // 320000 GEMM rows
#define ROWELTS (ROWS * CH)        // 10,240,000 floats per feature buffer

typedef float v2f __attribute__((ext_vector_type(2)));
typedef float v8f __attribute__((ext_vector_type(8)));

// ---------------------------------------------------------------------------
// x (B,N,C,T) -> h0 (BT,N,C), bt = b*T + t. One (b,n) per 256-thread block.
// Coalesced global read of 256 floats -> LDS (padded) -> coalesced write:
// each wave writes one contiguous 32-float run of h0.
// ---------------------------------------------------------------------------
__global__ void k_transpose(const float* __restrict__ x, float* __restrict__ h0) {
    __shared__ float s[CH * 9];          // stride-9 pad: conflict-free
    const int blk = blockIdx.x;          // b*N + n
    const int b   = blk / NNODES;
    const int n   = blk - b * NNODES;
    const int tid = threadIdx.x;         // 256 = C*T
    // read: flat (c,t) order matches x layout
    const int c = tid >> 3, t = tid & 7;
    s[c * 9 + t] = x[(size_t)blk * (CH * TDIM) + tid];
    __syncthreads();
    const int c2 = tid & 31, t2 = tid >> 5;
    h0[((size_t)(b * TDIM + t2) * NNODES + n) * CH + c2] = s[c2 * 9 + t2];
}

// ---------------------------------------------------------------------------
// agg[bt, dst[e], c] += h[bt, src[e], c].  One lane per (e, bt, c):
// a wave covers all 32 channels of one (edge, slice) -> contiguous 128B
// load + contiguous atomic adds; 32 consecutive waves share one edge.
// ---------------------------------------------------------------------------
__global__ void k_scatter(const float* __restrict__ h,
                          const long long* __restrict__ ei,
                          float* __restrict__ agg) {
    const int gt = blockIdx.x * 256 + threadIdx.x;
    const int c  = gt & 31;
    const int w  = gt >> 5;
    const int bt = w & 31;
    const int e  = w >> 5;
    const int src = (int)ei[e];
    const int dst = (int)ei[NEDGE + e];
    const float v = h[(bt * NNODES + src) * CH + c];
    atomicAdd(&agg[(bt * NNODES + dst) * CH + c], v);
}

// ---------------------------------------------------------------------------
// out = (h + agg) @ W + bias using V_WMMA_F32_16X16X4_F32.
// Block = 256 threads = 8 waves; each wave owns a 16-row x 32-col tile.
// Rows are the flat (bt,n) index; 128 rows per block, cooperatively staged
// in LDS (stride-33 pad).  K = 32 -> 8 WMMA steps of K=4, two N-halves.
// Accumulators are bias-initialized.  FINAL: fuse ReLU + (B,N,C,T) layout.
// ---------------------------------------------------------------------------
template <bool FINAL>
__global__ void k_gemm(const float* __restrict__ h, const float* __restrict__ agg,
                       const float* __restrict__ W, const float* __restrict__ bias,
                       float* __restrict__ out) {
    __shared__ float sM[128 * 33];
    __shared__ float sB[32 * 32];
    __shared__ float sBias[32];

    const int tid  = threadIdx.x;
    const int base = blockIdx.x * (128 * CH);

    // Stage M = h + agg (128 rows x 32 cols), 16B-vector global loads.
    const float4* h4 = (const float4*)(h + base);
    const float4* a4 = (const float4*)(agg + base);
#pragma unroll
    for (int i4 = tid; i4 < 1024; i4 += 256) {
        float4 hv = h4[i4];
        float4 av = a4[i4];
        const int row = i4 >> 3;
        const int col = (i4 & 7) * 4;
        float* d = &sM[row * 33 + col];
        d[0] = hv.x + av.x; d[1] = hv.y + av.y;
        d[2] = hv.z + av.z; d[3] = hv.w + av.w;
    }
#pragma unroll
    for (int i = tid; i < 1024; i += 256) sB[i] = W[i];
    if (tid < 32) sBias[tid] = bias[tid];
    __syncthreads();

    const int lane = tid & 31;
    const int wv   = tid >> 5;        // wave in block: 0..7
    const int wRow = wv * 16;         // tile's first row in sM
    const int l15  = lane & 15;
    const int lh   = lane >> 4;       // lane half selects K sub-pair

    v8f acc0, acc1;                   // cols 0-15 / 16-31
#pragma unroll
    for (int j = 0; j < 8; ++j) { acc0[j] = sBias[l15]; acc1[j] = sBias[l15 + 16]; }

#pragma unroll
    for (int kk = 0; kk < 8; ++kk) {
        const int kb   = kk * 4 + lh * 2;          // this lane-half's K base
        const int arow = wRow + l15;
        v2f a;  a.x  = sM[arow * 33 + kb];     a.y  = sM[arow * 33 + kb + 1];
        v2f b0; b0.x = sB[(kb + 0) * 32 + l15];      b0.y = sB[(kb + 1) * 32 + l15];
        v2f b1; b1.x = sB[(kb + 0) * 32 + l15 + 16]; b1.y = sB[(kb + 1) * 32 + l15 + 16];
        acc0 = __builtin_amdgcn_wmma_f32_16x16x4_f32(false, a, false, b0, (short)0, acc0, false, false);
        acc1 = __builtin_amdgcn_wmma_f32_16x16x4_f32(false, a, false, b1, (short)0, acc1, false, false);
    }

    const int rbase = blockIdx.x * 128 + wRow;
#pragma unroll
    for (int j = 0; j < 8; ++j) {
        const int r = rbase + j + 8 * lh;          // C/D layout: VGPR j -> rows j, j+8
        if (!FINAL) {
            out[(size_t)r * CH + l15]      = acc0[j];
            out[(size_t)r * CH + l15 + 16] = acc1[j];
        } else {
            const int bt = r / NNODES;
            const int n  = r - bt * NNODES;
            const int b  = bt >> 3, t = bt & 7;
            const float v0 = acc0[j] > 0.f ? acc0[j] : 0.f;
            const float v1 = acc1[j] > 0.f ? acc1[j] : 0.f;
            const size_t ob = (size_t)(b * NNODES + n) * CH;
            out[(ob + l15)      * TDIM + t] = v0;
            out[(ob + l15 + 16) * TDIM + t] = v1;
        }
    }
}

// ---------------------------------------------------------------------------
extern "C" void kernel_launch(void* const* d_in, const int* in_sizes, int n_in,
                              void* d_out, int out_size, void* d_ws, size_t ws_size,
                              hipStream_t stream) {
    const float*     x  = (const float*)d_in[0];
    const long long* ei = (const long long*)d_in[1];   // int64 edge_index (2,E)
    const float*     W1 = (const float*)d_in[2];
    const float*     b1 = (const float*)d_in[3];
    const float*     W2 = (const float*)d_in[4];
    const float*     b2 = (const float*)d_in[5];
    float* out = (float*)d_out;

    float* h0  = (float*)d_ws;         // 40.96 MB
    float* agg = h0 + ROWELTS;         // 40.96 MB

    const size_t aggBytes = sizeof(float) * (size_t)ROWELTS;
    const int scatterBlocks = (NEDGE * BT * 32) / 256;   // 640000

    // Layer 1
    hipMemsetAsync(agg, 0, aggBytes, stream);
    k_transpose<<<BDIM * NNODES, 256, 0, stream>>>(x, h0);
    k_scatter<<<scatterBlocks, 256, 0, stream>>>(h0, ei, agg);
    k_gemm<false><<<ROWS / 128, 256, 0, stream>>>(h0, agg, W1, b1, h0);  // in-place (row-local)

    // Layer 2 (+ ReLU + output transpose)
    hipMemsetAsync(agg, 0, aggBytes, stream);
    k_scatter<<<scatterBlocks, 256, 0, stream>>>(h0, ei, agg);
    k_gemm<true><<<ROWS / 128, 256, 0, stream>>>(h0, agg, W2, b2, out);
}